// GNN_82446192214266
// MI455X (gfx1250) — compile-verified
//
#include <hip/hip_runtime.h>

// GraphSAGE 2-layer, algebraically fused for MI455X (gfx1250, wave32).
// Pipeline:
//   K0: zero workspace (deg, agg1, agg2)
//   K1: edge pass 1 — scatter x[src] (9 f32) + degree into L2-resident accumulators
//   K2: node pass — WMMA f32 16x16x4: H=relu([agg1/deg | x] @ [W1l;W1r] + b1),
//       fused dots s=H·W2l, r=H·W2r (H never touches memory)
//   K3: edge pass 2 — scatter scalar s[src] per edge (linear map commuted
//       through the mean aggregation since layer-2 output dim == 1)
//   K4: out = sigmoid(agg2/deg + r + b2)

typedef __attribute__((ext_vector_type(2))) float v2f;
typedef __attribute__((ext_vector_type(8))) float v8f;

#define KDIM 20           // 18 features (9 agg + 9 x) padded to multiple of 4
#define WAVES_PER_BLOCK 4 // 128 threads, 4 wave32s, 64 nodes per block

__global__ void zero_kernel(float* __restrict__ p, int n) {
  int i = blockIdx.x * blockDim.x + threadIdx.x;
  if (i < n) p[i] = 0.0f;
}

__global__ void edge_pass1(const long long* __restrict__ src,
                           const long long* __restrict__ dst,
                           const float* __restrict__ x,
                           float* __restrict__ agg1,
                           float* __restrict__ deg, int E) {
  int e = blockIdx.x * blockDim.x + threadIdx.x;
  if (e >= E) return;
  int s = (int)src[e];
  int d = (int)dst[e];
  atomicAdd(&deg[d], 1.0f);
  const float* xs = x + (long)s * 9;
  float* ad = agg1 + (long)d * 9;
#pragma unroll
  for (int f = 0; f < 9; ++f) atomicAdd(&ad[f], xs[f]);
}

// Per-wave 16-node tile: H[16,256] = relu(A[16,20] @ W[20,256] + b1) via
// V_WMMA_F32_16X16X4_F32, then s = H.W2l, r = H.W2r reduced in-wave.
__global__ __launch_bounds__(128) void node_wmma(
    const float* __restrict__ x, const float* __restrict__ agg1,
    const float* __restrict__ deg, const float* __restrict__ W1l,
    const float* __restrict__ W1r, const float* __restrict__ b1,
    const float* __restrict__ W2l, const float* __restrict__ W2r,
    float* __restrict__ sArr, float* __restrict__ rArr, int N) {
  __shared__ float Wlds[KDIM][256];                 // [W1l ; W1r ; 0] stacked
  __shared__ float b1s[256], w2ls[256], w2rs[256];
  __shared__ float Asub[WAVES_PER_BLOCK][16][KDIM]; // per-wave A tiles

  const int tid = threadIdx.x;
  const int wave = tid >> 5;
  const int lane = tid & 31;
  const int base = (blockIdx.x * WAVES_PER_BLOCK + wave) * 16;

  // Stage stacked weight matrix (rows 0..8 = W1l, 9..17 = W1r, 18..19 = 0).
  for (int idx = tid; idx < KDIM * 256; idx += 128) {
    int k = idx >> 8, n = idx & 255;
    float v = 0.0f;
    if (k < 9)       v = W1l[k * 256 + n];
    else if (k < 18) v = W1r[(k - 9) * 256 + n];
    Wlds[k][n] = v;
  }
  for (int n = tid; n < 256; n += 128) {
    b1s[n]  = b1[n];
    w2ls[n] = W2l[n];
    w2rs[n] = W2r[n];
  }

  // Stage per-wave A tile: cols 0..8 = agg1/deg (mean), 9..17 = x, 18..19 = 0.
  for (int j = lane; j < 16 * KDIM; j += 32) {
    int row = j / KDIM, col = j % KDIM;
    int node = base + row;
    float v = 0.0f;
    if (node < N) {
      if (col < 9) {
        float inv = 1.0f / fmaxf(deg[node], 1.0f);
        v = agg1[(long)node * 9 + col] * inv;
      } else if (col < 18) {
        v = x[(long)node * 9 + (col - 9)];
      }
    }
    Asub[wave][row][col] = v;
  }
  __syncthreads();

  const int m  = lane & 15;  // row index within A half / col index within B
  const int hf = lane >> 4;  // lane half selects K pairs (ISA 32-bit A layout)

  float ps[8], pr[8];
#pragma unroll
  for (int i = 0; i < 8; ++i) { ps[i] = 0.0f; pr[i] = 0.0f; }

  // 16 N-tiles of 16 output columns each (256 total).
  for (int t = 0; t < 16; ++t) {
    const int ncol = t * 16 + m;
    const float wl = w2ls[ncol];
    const float wr = w2rs[ncol];

    v8f c;
    const float bias = b1s[ncol];   // bias folded into accumulator init
#pragma unroll
    for (int i = 0; i < 8; ++i) c[i] = bias;

#pragma unroll
    for (int k0 = 0; k0 < KDIM; k0 += 4) {
      v2f a, b;
      // A 16x4 f32: VGPR0 = K k0 (lanes 0-15) / k0+2 (16-31); VGPR1 = k0+1 / k0+3
      a.x = Asub[wave][m][k0 + 2 * hf];
      a.y = Asub[wave][m][k0 + 2 * hf + 1];
      // B 4x16 f32: mirrored layout, N striped across lanes
      b.x = Wlds[k0 + 2 * hf][ncol];
      b.y = Wlds[k0 + 2 * hf + 1][ncol];
      c = __builtin_amdgcn_wmma_f32_16x16x4_f32(
          /*neg_a=*/false, a, /*neg_b=*/false, b,
          /*c_mod=*/(short)0, c, /*reuse_a=*/false, /*reuse_b=*/false);
    }

    // ReLU + fused dot with W2l/W2r. D layout: VGPR i -> M = i (+8 in hi half).
#pragma unroll
    for (int i = 0; i < 8; ++i) {
      float h = c[i] > 0.0f ? c[i] : 0.0f;
      ps[i] += h * wl;
      pr[i] += h * wr;
    }
  }

  // Reduce each row's partial dot across its 16 lanes (xor stays in-half).
#pragma unroll
  for (int i = 0; i < 8; ++i) {
    float sv = ps[i], rv = pr[i];
#pragma unroll
    for (int off = 8; off >= 1; off >>= 1) {
      sv += __shfl_xor(sv, off, 32);
      rv += __shfl_xor(rv, off, 32);
    }
    int node = base + i + hf * 8;
    if (m == 0 && node < N) {
      sArr[node] = sv;
      rArr[node] = rv;
    }
  }
}

__global__ void edge_pass2(const long long* __restrict__ src,
                           const long long* __restrict__ dst,
                           const float* __restrict__ s,
                           float* __restrict__ agg2, int E) {
  int e = blockIdx.x * blockDim.x + threadIdx.x;
  if (e >= E) return;
  atomicAdd(&agg2[(int)dst[e]], s[(int)src[e]]);
}

__global__ void final_kernel(const float* __restrict__ agg2,
                             const float* __restrict__ deg,
                             const float* __restrict__ r,
                             const float* __restrict__ b2,
                             float* __restrict__ out, int N) {
  int n = blockIdx.x * blockDim.x + threadIdx.x;
  if (n >= N) return;
  float z = agg2[n] / fmaxf(deg[n], 1.0f) + r[n] + b2[0];
  out[n] = 1.0f / (1.0f + __expf(-z));
}

extern "C" void kernel_launch(void* const* d_in, const int* in_sizes, int n_in,
                              void* d_out, int out_size, void* d_ws, size_t ws_size,
                              hipStream_t stream) {
  const float*     x   = (const float*)d_in[0];
  const long long* ei  = (const long long*)d_in[1]; // int64 [2, E] row-major
  const float*     W1l = (const float*)d_in[2];
  const float*     W1r = (const float*)d_in[3];
  const float*     b1  = (const float*)d_in[4];
  const float*     W2l = (const float*)d_in[5];
  const float*     W2r = (const float*)d_in[6];
  const float*     b2  = (const float*)d_in[7];
  float* out = (float*)d_out;

  const int N = in_sizes[0] / 9;
  const int E = in_sizes[1] / 2;
  const long long* src = ei;
  const long long* dst = ei + E;

  // Workspace layout (floats): deg[N] | agg1[9N] | agg2[N] | s[N] | r[N]
  float* wf   = (float*)d_ws;
  float* deg  = wf;
  float* agg1 = wf + (size_t)N;
  float* agg2 = wf + (size_t)10 * N;
  float* sArr = wf + (size_t)11 * N;
  float* rArr = wf + (size_t)12 * N;

  const int nz = 11 * N; // deg + agg1 + agg2 need zeroing
  zero_kernel<<<(nz + 255) / 256, 256, 0, stream>>>(wf, nz);

  edge_pass1<<<(E + 255) / 256, 256, 0, stream>>>(src, dst, x, agg1, deg, E);

  const int nodeBlocks = (N + 16 * WAVES_PER_BLOCK - 1) / (16 * WAVES_PER_BLOCK);
  node_wmma<<<nodeBlocks, 128, 0, stream>>>(x, agg1, deg, W1l, W1r, b1,
                                            W2l, W2r, sArr, rArr, N);

  edge_pass2<<<(E + 255) / 256, 256, 0, stream>>>(src, dst, sArr, agg2, E);

  final_kernel<<<(N + 255) / 256, 256, 0, stream>>>(agg2, deg, rArr, b2, out, N);
}